// EdgeEncoding_73839077752851
// MI455X (gfx1250) — compile-verified
//
#include <hip/hip_runtime.h>
#include <hip/hip_bf16.h>

typedef __attribute__((ext_vector_type(2))) float v2f;
typedef __attribute__((ext_vector_type(8))) float v8f;

// Kernel A: dot[e][p] = sum_k emb[e][k] * evec[p][k]  via V_WMMA_F32_16X16X4_F32.
// One wave (32 lanes) computes a 16(E-rows) x 16(P-cols, cols 5..15 zero) tile,
// accumulating K=128 in steps of 4.
__global__ void edge_dot_wmma_kernel(const float* __restrict__ emb,   // (E,128)
                                     const float* __restrict__ evec,  // (5,128)
                                     float* __restrict__ dot,         // (E,5)
                                     int E)
{
    const int gtid = blockIdx.x * blockDim.x + threadIdx.x;
    const int wave = gtid >> 5;          // one 16-row tile per wave
    const int lane = threadIdx.x & 31;
    const int m    = lane & 15;          // A: row within tile / B,D: column N
    const int kh   = lane >> 4;          // K half-select (0 -> K{0,1}, 1 -> K{2,3})
    const int row0 = wave * 16;
    if (row0 >= E) return;               // wave-uniform: EXEC stays all-1s for WMMA

    // A fragment source: emb[row0+m][k + 2*kh], contiguous pair
    const float* __restrict__ arow = emb + (size_t)(row0 + m) * 128 + 2 * kh;
    // B fragment source: column n = m of B (= evec[n][k]); pad n>=5 with zeros
    // via multiply-mask so no divergent branches touch the WMMA loop.
    const int   nclamp = (m < 5) ? m : 4;
    const float nmask  = (m < 5) ? 1.0f : 0.0f;
    const float* __restrict__ bcol = evec + (size_t)nclamp * 128 + 2 * kh;

    v8f acc = {};
    #pragma unroll 4
    for (int k = 0; k < 128; k += 4) {
        v2f a, b;
        a.x = arow[k];
        a.y = arow[k + 1];
        b.x = bcol[k]     * nmask;
        b.y = bcol[k + 1] * nmask;
        // (neg_a, A, neg_b, B, c_mod, C, reuse_a, reuse_b)
        acc = __builtin_amdgcn_wmma_f32_16x16x4_f32(
            false, a, false, b, (short)0, acc, false, false);
    }

    // D layout: acc[r] holds element (M = r + 8*kh, N = m). Store only N<5.
    if (m < 5) {
        const int ebase = row0 + 8 * kh;
        #pragma unroll
        for (int r = 0; r < 8; ++r) {
            dot[(size_t)(ebase + r) * 5 + m] = acc[r];
        }
    }
}

// Kernel B: per (i,j) pair, sum the <=5 precomputed dots selected by
// edge_paths, divide by the count of valid entries (0 -> 0). The 160 KB dot
// table is L2-resident; edge_paths streams at full HBM bandwidth.
__global__ void edge_enc_gather_kernel(const int* __restrict__ paths,  // (N*N,5)
                                       const float* __restrict__ dot,  // (E,5)
                                       float* __restrict__ out,        // (N*N)
                                       int total)
{
    const int t = blockIdx.x * blockDim.x + threadIdx.x;
    if (t >= total) return;

    const int* __restrict__ p = paths + (size_t)t * 5;
    float s = 0.0f;
    int cnt = 0;
    #pragma unroll
    for (int q = 0; q < 5; ++q) {
        const int e = p[q];
        if (e >= 0) {
            s += dot[(size_t)e * 5 + q];
            ++cnt;
        }
    }
    out[t] = (cnt != 0) ? (s / (float)cnt) : 0.0f;
}

extern "C" void kernel_launch(void* const* d_in, const int* in_sizes, int n_in,
                              void* d_out, int out_size, void* d_ws, size_t ws_size,
                              hipStream_t stream) {
    // setup_inputs order: x (unused), edge_embedding, edge_paths, edge_vector
    const float* emb   = (const float*)d_in[1];
    const int*   paths = (const int*)  d_in[2];
    const float* evec  = (const float*)d_in[3];
    float*       out   = (float*)d_out;
    float*       dot   = (float*)d_ws;          // E*5 floats = 160 KB scratch

    const int E     = in_sizes[1] / 128;        // 8192
    const int total = in_sizes[2] / 5;          // N*N = 262144

    // Kernel A: one wave per 16-row tile; 4 waves (128 threads) per block.
    const int tiles  = (E + 15) / 16;           // 512 waves
    const int blocksA = (tiles + 3) / 4;        // 128 blocks of 128 threads
    edge_dot_wmma_kernel<<<blocksA, 128, 0, stream>>>(emb, evec, dot, E);

    // Kernel B: one thread per (i,j).
    const int blocksB = (total + 255) / 256;
    edge_enc_gather_kernel<<<blocksB, 256, 0, stream>>>(paths, dot, out, total);
}